// SimCLR_clear_loss_14688788152461
// MI455X (gfx1250) — compile-verified
//
#include <hip/hip_runtime.h>
#include <math.h>

// ---------------------------------------------------------------------------
// SimCLR supervised-contrastive loss, MI455X (gfx1250), wave32.
//   denom_i = sum_{j: lab_j==lab_i, j!=i} exp(<out_i,out_j>/T)
//   loss    = mean_i( log(denom_i) - <out1_p,out2_p>/T )
// GEMM via V_WMMA_F32_16X16X4_F32; 2x2 register blocking (32x32 per wave,
// 1 b64 load per WMMA); symmetric super-blocks computed once.
// ---------------------------------------------------------------------------

typedef __attribute__((ext_vector_type(2))) float v2f;
typedef __attribute__((ext_vector_type(4))) float v4f;
typedef __attribute__((ext_vector_type(8))) float v8f;

#define DIMK   512
#define NROWS  8192      // 2B
#define BHALF  4096      // B
#define NB     256       // NROWS/32 super-blocks per matrix dim
#define INV_T  2.0f      // 1/TEMPERATURE

// ---- zero the workspace (denom[NROWS] + scalar accumulator) ----------------
__global__ void simclr_zero_ws(float* __restrict__ ws) {
    int i = blockIdx.x * blockDim.x + threadIdx.x;
    if (i < NROWS + 1) ws[i] = 0.0f;
}

// ---- epilogue for one 16x16 sub-tile: exp + mask + row/col reduction -------
// C/D layout: lane L, vgpr r -> M = r + 8*(L>>4), N = L & 15
__device__ __forceinline__
void tile_epilogue(const v8f& acc, int it, int jt, bool addcol,
                   const int* __restrict__ labels,
                   float* __restrict__ denom, int lane) {
    const int m    = lane & 15;
    const int half = lane >> 4;
    const int gj   = jt * 16 + m;        // global column index for this lane
    const int labj = labels[gj];

    float colsum = 0.0f;
#pragma unroll
    for (int r = 0; r < 8; ++r) {
        const int gi    = it * 16 + r + 8 * half;
        const bool keep = (labels[gi] == labj) & (gi != gj);
        const float ev  = keep ? __expf(acc[r] * INV_T) : 0.0f;

        // row sum across the 16 lanes of this half
        float s = ev;
        s += __shfl_xor(s, 1, 32);
        s += __shfl_xor(s, 2, 32);
        s += __shfl_xor(s, 4, 32);
        s += __shfl_xor(s, 8, 32);
        if (m == 0) atomicAdd(&denom[gi], s);   // lanes 0,16 -> rows r, r+8

        colsum += ev;
    }
    // mirrored contribution (off-diagonal super-blocks only)
    if (addcol) {
        colsum += __shfl_xor(colsum, 16, 32);   // combine both M-halves
        if (half == 0) atomicAdd(&denom[gj], colsum);
    }
}

// ---- main kernel: fused GEMM + exp + mask + reductions ---------------------
// grid = (NB/8, NB), block = 256 (8 waves). Wave w handles 32x32 super-block
// (ib = blockIdx.y, jb = blockIdx.x*8 + w), upper triangle only.
__global__ __launch_bounds__(256)
void simclr_denom(const float* __restrict__ out,
                  const int*   __restrict__ labels,
                  float*       __restrict__ denom) {
    const int wave = threadIdx.x >> 5;
    const int lane = threadIdx.x & 31;
    const int jb = blockIdx.x * 8 + wave;
    const int ib = blockIdx.y;
    if (jb < ib) return;                 // wave-uniform: EXEC stays all-ones

    const int m    = lane & 15;          // row-in-tile (A) / col-in-tile (B)
    const int half = lane >> 4;          // K sub-slab selector (ISA layout)

    // f32 16x4 A / 4x16 B lane view: out[row][k + 2*half + {0,1}]
    const float* __restrict__ A0 = out + (size_t)(ib * 32 + m)      * DIMK + 2 * half;
    const float* __restrict__ A1 = out + (size_t)(ib * 32 + 16 + m) * DIMK + 2 * half;
    const float* __restrict__ B0 = out + (size_t)(jb * 32 + m)      * DIMK + 2 * half;
    const float* __restrict__ B1 = out + (size_t)(jb * 32 + 16 + m) * DIMK + 2 * half;

    v8f acc00 = {}, acc01 = {}, acc10 = {}, acc11 = {};
#pragma unroll 4
    for (int k = 0; k < DIMK; k += 4) {
        v2f a0 = *(const v2f*)(A0 + k);
        v2f a1 = *(const v2f*)(A1 + k);
        v2f b0 = *(const v2f*)(B0 + k);
        v2f b1 = *(const v2f*)(B1 + k);
        acc00 = __builtin_amdgcn_wmma_f32_16x16x4_f32(false, a0, false, b0,
                                                      (short)0, acc00, false, false);
        acc01 = __builtin_amdgcn_wmma_f32_16x16x4_f32(false, a0, false, b1,
                                                      (short)0, acc01, false, false);
        acc10 = __builtin_amdgcn_wmma_f32_16x16x4_f32(false, a1, false, b0,
                                                      (short)0, acc10, false, false);
        acc11 = __builtin_amdgcn_wmma_f32_16x16x4_f32(false, a1, false, b1,
                                                      (short)0, acc11, false, false);
    }

    const bool addcol = (ib != jb);      // diagonal block: full 32x32 computed,
                                         // row sums already cover everything
    tile_epilogue(acc00, ib * 2 + 0, jb * 2 + 0, addcol, labels, denom, lane);
    tile_epilogue(acc01, ib * 2 + 0, jb * 2 + 1, addcol, labels, denom, lane);
    tile_epilogue(acc10, ib * 2 + 1, jb * 2 + 0, addcol, labels, denom, lane);
    tile_epilogue(acc11, ib * 2 + 1, jb * 2 + 1, addcol, labels, denom, lane);
}

// ---- per-pair positive term + log(denom) reduction -------------------------
__global__ __launch_bounds__(256)
void simclr_loss(const float* __restrict__ o1,
                 const float* __restrict__ o2,
                 const float* __restrict__ denom,
                 float*       __restrict__ acc) {
    const int wave = (int)((blockIdx.x * blockDim.x + threadIdx.x) >> 5);
    const int lane = threadIdx.x & 31;
    if (wave >= BHALF) return;

    const float* __restrict__ a = o1 + (size_t)wave * DIMK + lane * 16;
    const float* __restrict__ b = o2 + (size_t)wave * DIMK + lane * 16;

    float d = 0.0f;
#pragma unroll
    for (int q = 0; q < 4; ++q) {
        v4f av = *(const v4f*)(a + q * 4);
        v4f bv = *(const v4f*)(b + q * 4);
        d += av[0] * bv[0] + av[1] * bv[1] + av[2] * bv[2] + av[3] * bv[3];
    }
    d += __shfl_xor(d, 1, 32);
    d += __shfl_xor(d, 2, 32);
    d += __shfl_xor(d, 4, 32);
    d += __shfl_xor(d, 8, 32);
    d += __shfl_xor(d, 16, 32);

    if (lane == 0) {
        const float term = __logf(denom[wave]) + __logf(denom[wave + BHALF])
                         - 2.0f * INV_T * d;
        atomicAdd(acc, term);
    }
}

// ---- finalize: mean over 2B rows -------------------------------------------
__global__ void simclr_finalize(const float* __restrict__ acc,
                                float* __restrict__ out) {
    out[0] = acc[0] * (1.0f / (float)NROWS);
}

// ---------------------------------------------------------------------------
extern "C" void kernel_launch(void* const* d_in, const int* in_sizes, int n_in,
                              void* d_out, int out_size, void* d_ws, size_t ws_size,
                              hipStream_t stream) {
    const float* out_m  = (const float*)d_in[0];   // [2B, D]
    const float* out_1  = (const float*)d_in[1];   // [B, D]
    const float* out_2  = (const float*)d_in[2];   // [B, D]
    const int*   labels = (const int*)d_in[3];     // [2B]

    float* ws    = (float*)d_ws;
    float* denom = ws;            // [NROWS]
    float* acc   = ws + NROWS;    // [1]

    simclr_zero_ws<<<(NROWS + 1 + 255) / 256, 256, 0, stream>>>(ws);

    dim3 grid(NB / 8, NB);        // 8 waves/block -> 8 super-blocks along j
    simclr_denom<<<grid, 256, 0, stream>>>(out_m, labels, denom);

    simclr_loss<<<(BHALF * 32) / 256, 256, 0, stream>>>(out_1, out_2, denom, acc);

    simclr_finalize<<<1, 1, 0, stream>>>(acc, (float*)d_out);
}